// Qwen35GatedDeltaNet_73933567034080
// MI455X (gfx1250) — compile-verified
//
#include <hip/hip_runtime.h>
#include <hip/hip_bf16.h>
#include <math.h>

typedef __attribute__((ext_vector_type(2))) float v2f;
typedef __attribute__((ext_vector_type(4))) float v4f;
typedef __attribute__((ext_vector_type(8))) float v8f;
typedef __attribute__((ext_vector_type(4))) int   i32x4;

#define HIDN 4096
#define NH   32
#define DKD  128
#define DVD  128
#define QKVN (NH*DKD*2 + NH*DVD)          /* 12288 */
#define ROWS1 (QKVN + HIDN + NH + NH)     /* 16448 */
#define SPITCH 132                        /* LDS row pitch for S (bank spread) */

__device__ __forceinline__ float wave_reduce_sum(float v) {
#pragma unroll
  for (int m = 16; m > 0; m >>= 1) v += __shfl_xor(v, m, 32);
  return v;
}

// ---- async global -> LDS b128 copy (ASYNCcnt path) ------------------------
__device__ __forceinline__ void async_copy_b128(const float* g, float* l) {
#if __has_builtin(__builtin_amdgcn_global_load_async_to_lds_b128)
  __attribute__((address_space(1))) i32x4* gp =
      (__attribute__((address_space(1))) i32x4*)(uintptr_t)g;
  __attribute__((address_space(3))) i32x4* lp =
      (__attribute__((address_space(3))) i32x4*)(unsigned)(uintptr_t)l;
  __builtin_amdgcn_global_load_async_to_lds_b128(gp, lp, 0, 0);
#else
  unsigned loff = (unsigned)(uintptr_t)l;   // low 32 bits = LDS byte offset
  asm volatile("global_load_async_to_lds_b128 %0, %1, off"
               :: "v"(loff), "v"(g) : "memory");
#endif
}

__device__ __forceinline__ void wait_async0() {
#if __has_builtin(__builtin_amdgcn_s_wait_asynccnt)
  __builtin_amdgcn_s_wait_asynccnt(0);
#else
  asm volatile("s_wait_asynccnt 0x0" ::: "memory");
#endif
}

// ---------------------------------------------------------------------------
// Kernel 1: fused GEMV  ws[row] = dot(W[row], hs)   for W_qkv|W_z|W_a|W_b
// One wave32 per row; hs staged in LDS via async b128; b128 streaming loads.
// ---------------------------------------------------------------------------
__global__ __launch_bounds__(256) void gemv_fused(
    const float* __restrict__ Wqkv, const float* __restrict__ Wz,
    const float* __restrict__ Wa,   const float* __restrict__ Wb,
    const float* __restrict__ hs,   float* __restrict__ ws)
{
  __shared__ __align__(16) float shs[HIDN];
  const int tid = threadIdx.x;
  for (int i = tid * 4; i < HIDN; i += 1024) async_copy_b128(hs + i, shs + i);
  wait_async0();
  __syncthreads();

  const int wave = tid >> 5, lane = tid & 31;
  const int row  = blockIdx.x * 8 + wave;
  if (row >= ROWS1) return;

  const float* Wrow;
  if      (row < QKVN)             Wrow = Wqkv + (size_t)row * HIDN;
  else if (row < QKVN + HIDN)      Wrow = Wz   + (size_t)(row - QKVN) * HIDN;
  else if (row < QKVN + HIDN + NH) Wrow = Wa   + (size_t)(row - QKVN - HIDN) * HIDN;
  else                             Wrow = Wb   + (size_t)(row - QKVN - HIDN - NH) * HIDN;

  float acc = 0.f;
  for (int i = lane * 4; i < HIDN; i += 128) {
    __builtin_prefetch(Wrow + i + 1024, 0, 1);   // global_prefetch_b8
    v4f w = *(const v4f*)(Wrow + i);
    v4f h = *(const v4f*)(shs + i);
    acc = fmaf(w.x, h.x, acc); acc = fmaf(w.y, h.y, acc);
    acc = fmaf(w.z, h.z, acc); acc = fmaf(w.w, h.w, acc);
  }
  acc = wave_reduce_sum(acc);
  if (lane == 0) ws[row] = acc;
}

// ---------------------------------------------------------------------------
// Kernel 2: per-head conv + activation + gated-delta state update.
// S (64KB) is staged into LDS with async b128 copies, overlapped with the
// conv/normalization phase. S@k and S@q via V_WMMA_F32_16X16X4_F32
// (B broadcast across N). S' = g*S + delta (x) k streamed LDS -> global.
// ---------------------------------------------------------------------------
__global__ __launch_bounds__(256) void head_update(
    const float* __restrict__ Sin,     // delta_state, [h][v][k]
    const float* __restrict__ cstate,  // conv_state  [QKV][3]
    const float* __restrict__ cw,      // conv_w      [QKV][4]
    const float* __restrict__ A_log, const float* __restrict__ dt_bias,
    const float* __restrict__ rms_w,
    const float* __restrict__ ws,      // qkv | z | a | b
    float* __restrict__ outvec,        // [H*DV] to kernel 3
    float* __restrict__ Sout,          // d_out + HID
    float* __restrict__ ncs)           // d_out + HID + H*DV*DK
{
  const int h    = blockIdx.x;
  const int tid  = threadIdx.x;
  const int lane = tid & 31, wv = tid >> 5;

  __shared__ __align__(16) float sS[DVD * SPITCH];  // padded S tile
  __shared__ __align__(16) float qb[DKD];
  __shared__ __align__(16) float kb[DKD];
  __shared__ __align__(16) float vb[DVD];
  __shared__ float r0s[DVD], r1s[DVD], dls[DVD], outs[DVD];
  __shared__ float red[8];

  // ---- kick off async S -> LDS (16 x b128 per thread), overlapped below
  const float* Sh = Sin + (size_t)h * DVD * DKD;
  for (int i = tid * 4; i < DVD * DKD; i += 1024) {
    int vrow = i >> 7, kcol = i & 127;
    async_copy_b128(Sh + i, sS + vrow * SPITCH + kcol);
  }

  const float* qkv = ws;
  const float* zv  = ws + QKVN;
  const float* av  = ws + QKVN + HIDN;
  const float* bv  = ws + QKVN + HIDN + NH;

  // ---- conv (depth 4) + SiLU for this head's q,k,v rows; emit new_conv_state
  for (int e = tid; e < 3 * DKD; e += 256) {
    int seg = e >> 7, d = e & 127;
    int gi  = seg * (NH * DKD) + h * DKD + d;
    float x  = qkv[gi];
    float c0 = cstate[gi*3+0], c1 = cstate[gi*3+1], c2 = cstate[gi*3+2];
    float w0 = cw[gi*4+0], w1 = cw[gi*4+1], w2 = cw[gi*4+2], w3 = cw[gi*4+3];
    float co = fmaf(w3, x, fmaf(w2, c2, fmaf(w1, c1, w0*c0)));
    float act = co / (1.f + __expf(-co));
    if (seg == 0) qb[d] = act; else if (seg == 1) kb[d] = act; else vb[d] = act;
    ncs[gi*3+0] = c1; ncs[gi*3+1] = c2; ncs[gi*3+2] = x;
  }
  __syncthreads();

  // ---- L2 normalization of q (incl. 1/sqrt(DK)) and k
  float pq = (tid < DKD) ? qb[tid]*qb[tid] : 0.f;
  float pk = (tid < DKD) ? kb[tid]*kb[tid] : 0.f;
  pq = wave_reduce_sum(pq); pk = wave_reduce_sum(pk);
  if (lane == 0) red[wv] = pq;
  __syncthreads();
  float sq = red[0]+red[1]+red[2]+red[3]+red[4]+red[5]+red[6]+red[7];
  __syncthreads();
  if (lane == 0) red[wv] = pk;
  __syncthreads();
  float sk = red[0]+red[1]+red[2]+red[3]+red[4]+red[5]+red[6]+red[7];
  __syncthreads();
  float qscale = rsqrtf(sq + 1e-6f) * 0.08838834764831845f; // 1/sqrt(128)
  float kscale = rsqrtf(sk + 1e-6f);
  if (tid < DKD) { qb[tid] *= qscale; kb[tid] *= kscale; }
  __syncthreads();

  float pkq = (tid < DKD) ? qb[tid]*kb[tid] : 0.f;
  pkq = wave_reduce_sum(pkq);
  if (lane == 0) red[wv] = pkq;
  __syncthreads();
  float kq = red[0]+red[1]+red[2]+red[3]+red[4]+red[5]+red[6]+red[7];

  // ---- gate / beta scalars (redundant per thread)
  float xsp = av[h] + dt_bias[h];
  float sp  = (xsp > 20.f) ? xsp : log1pf(__expf(xsp));
  float g   = __expf(-__expf(A_log[h]) * sp);
  float beta = 1.f / (1.f + __expf(-bv[h]));

  // ---- S fully staged before pass 1
  wait_async0();
  __syncthreads();

  // ---- pass 1: r0 = S@k, r1 = S@q via f32 WMMA (wave wv owns rows wv*16..+15)
  const int m = lane & 15, kh = lane >> 4;
  const float* Srow = sS + (wv * 16 + m) * SPITCH;
  v8f c0 = {}; v8f c1 = {};
  for (int kc = 0; kc < DKD / 4; ++kc) {
    int col = kc * 4 + kh * 2;
    v2f a;  a.x  = Srow[col]; a.y  = Srow[col + 1];   // A 16x4 (ISA layout)
    v2f bk; bk.x = kb[col];   bk.y = kb[col + 1];     // B broadcast over N
    v2f bq; bq.x = qb[col];   bq.y = qb[col + 1];
    c0 = __builtin_amdgcn_wmma_f32_16x16x4_f32(false, a, false, bk, (short)0, c0, false, false);
    c1 = __builtin_amdgcn_wmma_f32_16x16x4_f32(false, a, false, bq, (short)0, c1, false, false);
  }
  // D layout: lane holds rows M = j + 8*(lane>>4); every N column identical.
  if ((lane & 15) == 0) {
    int base = wv * 16 + kh * 8;
#pragma unroll
    for (int j = 0; j < 8; ++j) { r0s[base + j] = c0[j]; r1s[base + j] = c1[j]; }
  }
  __syncthreads();

  // ---- delta, out = g*(S@q) + delta*(k.q)
  if (tid < DVD) {
    float retr = g * r0s[tid];
    float dl   = beta * (vb[tid] - retr);
    dls[tid]  = dl;
    outs[tid] = fmaf(dl, kq, g * r1s[tid]);
  }
  __syncthreads();

  // ---- RMS norm + silu(z) gate
  float po = (tid < DVD) ? outs[tid]*outs[tid] : 0.f;
  po = wave_reduce_sum(po);
  if (lane == 0) red[wv] = po;
  __syncthreads();
  float var = (red[0]+red[1]+red[2]+red[3]+red[4]+red[5]+red[6]+red[7]) * (1.f / DVD);
  __syncthreads();
  if (tid < DVD) {
    float o  = outs[tid] * rsqrtf(var + 1e-6f) * rms_w[tid];
    float zz = zv[h * DVD + tid];
    outvec[h * DVD + tid] = o * (zz / (1.f + __expf(-zz)));
  }

  // ---- pass 2: S' = g*S + delta (x) k, LDS b128 in / global b128 out
  float* So = Sout + (size_t)h * DVD * DKD;
  for (int i = tid * 4; i < DVD * DKD; i += 1024) {
    int vrow = i >> 7, kcol = i & 127;
    v4f s  = *(const v4f*)(sS + vrow * SPITCH + kcol);
    v4f kk = *(const v4f*)(kb + kcol);
    float dl = dls[vrow];
    v4f r;
    r.x = fmaf(dl, kk.x, g * s.x);
    r.y = fmaf(dl, kk.y, g * s.y);
    r.z = fmaf(dl, kk.z, g * s.z);
    r.w = fmaf(dl, kk.w, g * s.w);
    *(v4f*)(So + i) = r;
  }
}

// ---------------------------------------------------------------------------
// Kernel 3: result = W_out @ outvec
// ---------------------------------------------------------------------------
__global__ __launch_bounds__(256) void gemv_out(
    const float* __restrict__ Wout, const float* __restrict__ xin,
    float* __restrict__ out)
{
  __shared__ __align__(16) float sx[HIDN];
  const int tid = threadIdx.x;
  for (int i = tid * 4; i < HIDN; i += 1024) async_copy_b128(xin + i, sx + i);
  wait_async0();
  __syncthreads();

  const int wave = tid >> 5, lane = tid & 31;
  const int row  = blockIdx.x * 8 + wave;
  if (row >= HIDN) return;
  const float* Wrow = Wout + (size_t)row * HIDN;

  float acc = 0.f;
  for (int i = lane * 4; i < HIDN; i += 128) {
    __builtin_prefetch(Wrow + i + 1024, 0, 1);
    v4f w = *(const v4f*)(Wrow + i);
    v4f h = *(const v4f*)(sx + i);
    acc = fmaf(w.x, h.x, acc); acc = fmaf(w.y, h.y, acc);
    acc = fmaf(w.z, h.z, acc); acc = fmaf(w.w, h.w, acc);
  }
  acc = wave_reduce_sum(acc);
  if (lane == 0) out[row] = acc;
}

// ---------------------------------------------------------------------------
extern "C" void kernel_launch(void* const* d_in, const int* in_sizes, int n_in,
                              void* d_out, int out_size, void* d_ws, size_t ws_size,
                              hipStream_t stream)
{
  const float* hs   = (const float*)d_in[0];
  const float* S    = (const float*)d_in[1];
  const float* cs   = (const float*)d_in[2];
  const float* Wqkv = (const float*)d_in[3];
  const float* Wz   = (const float*)d_in[4];
  const float* Wa   = (const float*)d_in[5];
  const float* Wb   = (const float*)d_in[6];
  const float* Wout = (const float*)d_in[7];
  const float* cw   = (const float*)d_in[8];
  const float* Alog = (const float*)d_in[9];
  const float* dtb  = (const float*)d_in[10];
  const float* rmsw = (const float*)d_in[11];

  float* out    = (float*)d_out;                 // [0,4096) result
  float* Sout   = out + HIDN;                    // [4096, 4096+524288) S'
  float* ncs    = Sout + NH * DVD * DKD;         // new_conv_state (QKV*3)
  float* ws     = (float*)d_ws;                  // qkv|z|a|b
  float* outvec = ws + ROWS1;                    // gated per-head output (4096)

  gemv_fused<<<ROWS1 / 8, 256, 0, stream>>>(Wqkv, Wz, Wa, Wb, hs, ws);
  head_update<<<NH, 256, 0, stream>>>(S, cs, cw, Alog, dtb, rmsw, ws, outvec, Sout, ncs);
  gemv_out<<<HIDN / 8, 256, 0, stream>>>(Wout, outvec, out);
}